// GNNBackbone_63917703299286
// MI455X (gfx1250) — compile-verified
//
#include <hip/hip_runtime.h>
#include <hip/hip_bf16.h>

// ---------------------------------------------------------------------------
// GNN backbone (2x GCN + JK-concat readout) for MI455X / gfx1250.
// Dense GEMMs use native fp32 WMMA (v_wmma_f32_16x16x4_f32) -> exact fp32
// semantics; edge gather/scatter uses one wave32 per edge with float4 loads
// and global_atomic_add_f32 (working set ~100MB, fits the 192MB L2).
// ---------------------------------------------------------------------------

#define IN_DIM 128
#define HID    128
#define OUT_DIM 64
#define EPS_W  1e-8f

typedef float v2f __attribute__((ext_vector_type(2)));
typedef float v8f __attribute__((ext_vector_type(8)));

#if defined(__gfx1250__) && __has_builtin(__builtin_amdgcn_wmma_f32_16x16x4_f32)
#define HAVE_WMMA_F32 1
#endif

// ---------------------------------------------------------------- fill utils
__global__ void fill_f32_kernel(float* __restrict__ p, float v, long n) {
    long i = (long)blockIdx.x * blockDim.x + threadIdx.x;
    if (i < n) p[i] = v;
}

__global__ void init_minmax_kernel(unsigned* __restrict__ mm) {
    if (threadIdx.x == 0) {
        mm[0] = 0x7F800000u;  // +inf bits (min slot)
        mm[1] = 0x00000000u;  // 0 bits (max slot); means are >= 0
    }
}

// ------------------------------------------------- edge mean + global min/max
// mean over 4 attrs per edge; block-reduced bitwise atomic min/max (valid for
// non-negative floats: IEEE bits order == value order).
__global__ void edge_mean_minmax_kernel(const float* __restrict__ ea,
                                        float* __restrict__ ew,
                                        unsigned* __restrict__ mm, long E) {
    __shared__ unsigned smin[256];
    __shared__ unsigned smax[256];
    int tid = threadIdx.x;
    long e = (long)blockIdx.x * blockDim.x + tid;
    unsigned bmin = 0x7F800000u, bmax = 0u;
    if (e < E) {
        const float4 a = *(const float4*)(ea + e * 4);
        float m = 0.25f * (a.x + a.y + a.z + a.w);
        ew[e] = m;
        unsigned b = __float_as_uint(m);
        bmin = b; bmax = b;
    }
    smin[tid] = bmin; smax[tid] = bmax;
    __syncthreads();
    for (int s = 128; s > 0; s >>= 1) {
        if (tid < s) {
            smin[tid] = min(smin[tid], smin[tid + s]);
            smax[tid] = max(smax[tid], smax[tid + s]);
        }
        __syncthreads();
    }
    if (tid == 0) {
        atomicMin(&mm[0], smin[0]);
        atomicMax(&mm[1], smax[0]);
    }
}

// -------------------------------------- edge weight + degree accumulation
__global__ void edge_weight_deg_kernel(float* __restrict__ ew,
                                       const int* __restrict__ col,
                                       float* __restrict__ deg,
                                       const unsigned* __restrict__ mm, long E) {
    long e = (long)blockIdx.x * blockDim.x + threadIdx.x;
    if (e >= E) return;
    float mn = __uint_as_float(mm[0]);
    float mx = __uint_as_float(mm[1]);
    float m = ew[e];
    float range = mx - mn;
    float w = (range > EPS_W) ? (1.0f - (m - mn) / (range + EPS_W)) : 1.0f;
    w = fmaxf(w, 0.0f);
    ew[e] = w;
    atomicAdd(&deg[col[e]], w);
}

__global__ void rsqrt_kernel(float* __restrict__ deg, long n) {
    long i = (long)blockIdx.x * blockDim.x + threadIdx.x;
    if (i < n) deg[i] = rsqrtf(deg[i]);  // deg >= 1
}

__global__ void edge_norm_kernel(const float* __restrict__ ew,
                                 const int* __restrict__ row,
                                 const int* __restrict__ col,
                                 const float* __restrict__ dis,
                                 float* __restrict__ nrm, long E) {
    long e = (long)blockIdx.x * blockDim.x + threadIdx.x;
    if (e >= E) return;
    nrm[e] = dis[row[e]] * ew[e] * dis[col[e]];
}

// ---------------------------------------------------------------- transpose
// dst[c*R + r] = src[r*C + c]   (tiny weight matrices; once per launch)
__global__ void transpose_kernel(float* __restrict__ dst,
                                 const float* __restrict__ src, int R, int C) {
    long i = (long)blockIdx.x * blockDim.x + threadIdx.x;
    if (i >= (long)R * C) return;
    int r = (int)(i / C), c = (int)(i % C);
    dst[(size_t)c * R + r] = src[i];
}

// ----------------------------------------------- WMMA GEMM: C[N,128]=A@Bt
// A: [N,128] row-major. Bt: [128,128] row-major (pre-transposed weight).
// One wave per 16-row strip; 8 column tiles of 16x16 f32 accumulators.
// fp32 WMMA fragment layout (ISA 7.12.2):
//   A 16x4 : lanes 0-15 hold K={k0,k0+1}, lanes 16-31 hold K={k0+2,k0+3}
//   B 4x16 : VGPR0 = rows k0 / k0+2, VGPR1 = rows k0+1 / k0+3
__global__ void __launch_bounds__(256)
gemm_nt128_kernel(const float* __restrict__ A, const float* __restrict__ Bt,
                  float* __restrict__ C, int N) {
    const int wave = threadIdx.x >> 5;
    const int lane = threadIdx.x & 31;
    const int half = lane >> 4;
    const int lm   = lane & 15;
    const int m0   = (blockIdx.x * 8 + wave) * 16;
    if (m0 >= N) return;  // wave-uniform: EXEC stays all-1s for live waves

    int rowA = m0 + lm;
    if (rowA >= N) rowA = N - 1;  // clamp (stores are masked later)
    const float* __restrict__ arow = A + (size_t)rowA * 128;

#if defined(HAVE_WMMA_F32)
    v8f acc[8];
#pragma unroll
    for (int t = 0; t < 8; ++t)
        acc[t] = (v8f){0.f, 0.f, 0.f, 0.f, 0.f, 0.f, 0.f, 0.f};

    for (int k0 = 0; k0 < 128; k0 += 4) {
        const int ka = k0 + 2 * half;
        v2f a;
        a.x = arow[ka];
        a.y = arow[ka + 1];
        const float* __restrict__ b0 = Bt + (size_t)ka * 128 + lm;
        const float* __restrict__ b1 = Bt + (size_t)(ka + 1) * 128 + lm;
#pragma unroll
        for (int t = 0; t < 8; ++t) {
            v2f b;
            b.x = b0[t * 16];
            b.y = b1[t * 16];
            acc[t] = __builtin_amdgcn_wmma_f32_16x16x4_f32(
                false, a, false, b, (short)0, acc[t], false, false);
        }
    }
#pragma unroll
    for (int t = 0; t < 8; ++t) {
        const int colv = t * 16 + lm;
#pragma unroll
        for (int v = 0; v < 8; ++v) {
            const int rowv = m0 + v + 8 * half;
            if (rowv < N) C[(size_t)rowv * 128 + colv] = acc[t][v];
        }
    }
#else
    // scalar fallback (host pass / missing builtin)
    for (int t = 0; t < 8; ++t) {
        const int colv = t * 16 + lm;
        for (int v = 0; v < 8; ++v) {
            const int rowv = m0 + v + 8 * half;
            if (rowv >= N) continue;
            float s = 0.f;
            for (int k = 0; k < 128; ++k)
                s += A[(size_t)rowv * 128 + k] * Bt[(size_t)k * 128 + colv];
            C[(size_t)rowv * 128 + colv] = s;
        }
    }
#endif
}

// ------------------------------------- WMMA readout: out[N,64]=[h1|h2]@Wrt+br
// Wrt: [256,64] row-major. 4 column tiles, K=256 split across h1/h2.
__global__ void __launch_bounds__(256)
gemm_readout_kernel(const float* __restrict__ h1, const float* __restrict__ h2,
                    const float* __restrict__ Wrt, const float* __restrict__ br,
                    float* __restrict__ out, int N) {
    const int wave = threadIdx.x >> 5;
    const int lane = threadIdx.x & 31;
    const int half = lane >> 4;
    const int lm   = lane & 15;
    const int m0   = (blockIdx.x * 8 + wave) * 16;
    if (m0 >= N) return;

    int rowA = m0 + lm;
    if (rowA >= N) rowA = N - 1;
    const float* __restrict__ r1 = h1 + (size_t)rowA * 128;
    const float* __restrict__ r2 = h2 + (size_t)rowA * 128;

#if defined(HAVE_WMMA_F32)
    v8f acc[4];
#pragma unroll
    for (int t = 0; t < 4; ++t)
        acc[t] = (v8f){0.f, 0.f, 0.f, 0.f, 0.f, 0.f, 0.f, 0.f};

    for (int k0 = 0; k0 < 256; k0 += 4) {
        const int ka = k0 + 2 * half;
        v2f a;
        if (k0 < 128) { a.x = r1[ka];       a.y = r1[ka + 1]; }
        else          { a.x = r2[ka - 128]; a.y = r2[ka - 127]; }
        const float* __restrict__ b0 = Wrt + (size_t)ka * 64 + lm;
        const float* __restrict__ b1 = Wrt + (size_t)(ka + 1) * 64 + lm;
#pragma unroll
        for (int t = 0; t < 4; ++t) {
            v2f b;
            b.x = b0[t * 16];
            b.y = b1[t * 16];
            acc[t] = __builtin_amdgcn_wmma_f32_16x16x4_f32(
                false, a, false, b, (short)0, acc[t], false, false);
        }
    }
#pragma unroll
    for (int t = 0; t < 4; ++t) {
        const int colv = t * 16 + lm;
        const float bias = br[colv];
#pragma unroll
        for (int v = 0; v < 8; ++v) {
            const int rowv = m0 + v + 8 * half;
            if (rowv < N) out[(size_t)rowv * 64 + colv] = acc[t][v] + bias;
        }
    }
#else
    for (int t = 0; t < 4; ++t) {
        const int colv = t * 16 + lm;
        for (int v = 0; v < 8; ++v) {
            const int rowv = m0 + v + 8 * half;
            if (rowv >= N) continue;
            float s = br[colv];
            for (int k = 0; k < 128; ++k) {
                s += h1[(size_t)rowv * 128 + k] * Wrt[(size_t)k * 64 + colv];
                s += h2[(size_t)rowv * 128 + k] * Wrt[(size_t)(k + 128) * 64 + colv];
            }
            out[(size_t)rowv * 64 + colv] = s;
        }
    }
#endif
}

// --------------------------------------------------- edge scatter-accumulate
// One wave32 per edge: float4 gather of hW[row] (512B coalesced per wave),
// scaled atomic adds into agg[col]. agg/hW (~100MB) stay resident in L2.
__global__ void __launch_bounds__(256)
scatter_edges_kernel(const float* __restrict__ hW, const float* __restrict__ nrm,
                     const int* __restrict__ row, const int* __restrict__ col,
                     float* __restrict__ agg, long E) {
    const int wave = threadIdx.x >> 5;
    const int lane = threadIdx.x & 31;
    const long e = (long)blockIdx.x * 8 + wave;
    if (e >= E) return;
    const int r = row[e];
    const int c = col[e];
    const float w = nrm[e];
    const float4 v = *(const float4*)(hW + (size_t)r * 128 + lane * 4);
    float* __restrict__ dst = agg + (size_t)c * 128 + lane * 4;
    atomicAdd(dst + 0, w * v.x);
    atomicAdd(dst + 1, w * v.y);
    atomicAdd(dst + 2, w * v.z);
    atomicAdd(dst + 3, w * v.w);
}

// ------------------------------------- h = relu(agg + dis^2 * hW + b)
__global__ void combine_relu_kernel(const float* __restrict__ agg,
                                    const float* __restrict__ hW,
                                    const float* __restrict__ dis,
                                    const float* __restrict__ bias,
                                    float* __restrict__ out, int N) {
    long i = (long)blockIdx.x * blockDim.x + threadIdx.x;  // over N*32 float4s
    if (i >= (long)N * 32) return;
    const int n = (int)(i >> 5);
    const int q = (int)(i & 31) * 4;
    const float d = dis[n];
    const float dd = d * d;
    const float4 a = *(const float4*)(agg + (size_t)n * 128 + q);
    const float4 h = *(const float4*)(hW + (size_t)n * 128 + q);
    const float4 b = *(const float4*)(bias + q);
    float4 r;
    r.x = fmaxf(a.x + dd * h.x + b.x, 0.f);
    r.y = fmaxf(a.y + dd * h.y + b.y, 0.f);
    r.z = fmaxf(a.z + dd * h.z + b.z, 0.f);
    r.w = fmaxf(a.w + dd * h.w + b.w, 0.f);
    *(float4*)(out + (size_t)n * 128 + q) = r;
}

// ---------------------------------------------------------------------------
extern "C" void kernel_launch(void* const* d_in, const int* in_sizes, int n_in,
                              void* d_out, int out_size, void* d_ws, size_t ws_size,
                              hipStream_t stream) {
    const float* x   = (const float*)d_in[0];
    const int*   ei  = (const int*)d_in[1];   // [2,E]: row then col
    const float* ea  = (const float*)d_in[2]; // [E,4]
    const float* W1  = (const float*)d_in[3];
    const float* b1  = (const float*)d_in[4];
    const float* W2  = (const float*)d_in[5];
    const float* b2  = (const float*)d_in[6];
    const float* Wr  = (const float*)d_in[7];
    const float* br  = (const float*)d_in[8];
    float* out = (float*)d_out;

    const long N = in_sizes[0] / IN_DIM;
    const long E = in_sizes[2] / 4;
    const int* row = ei;
    const int* col = ei + E;

    // ------- workspace carve-up (all fp32, 256B aligned)
    char* ws = (char*)d_ws;
    size_t off = 0;
    auto carve = [&](size_t bytes) -> char* {
        char* p = ws + off;
        off += (bytes + 255) & ~(size_t)255;
        return p;
    };
    float*    ew   = (float*)carve(E * 4);
    float*    nrm  = (float*)carve(E * 4);
    float*    deg  = (float*)carve(N * 4);          // becomes dis after rsqrt
    float*    hW   = (float*)carve(N * 128 * 4);
    float*    agg  = (float*)carve(N * 128 * 4);
    float*    h1   = (float*)carve(N * 128 * 4);
    float*    h2   = (float*)carve(N * 128 * 4);
    float*    Wt1  = (float*)carve(128 * 128 * 4);
    float*    Wt2  = (float*)carve(128 * 128 * 4);
    float*    Wrt  = (float*)carve(256 * 64 * 4);
    unsigned* mm   = (unsigned*)carve(2 * 4);
    (void)ws_size; (void)n_in; (void)out_size;

    const int BT = 256;
    auto blk1 = [&](long n) { return (unsigned)((n + BT - 1) / BT); };

    // ------- edge weights, degree, norm
    init_minmax_kernel<<<1, 32, 0, stream>>>(mm);
    fill_f32_kernel<<<blk1(N), BT, 0, stream>>>(deg, 1.0f, N);  // self-loop
    edge_mean_minmax_kernel<<<blk1(E), BT, 0, stream>>>(ea, ew, mm, E);
    edge_weight_deg_kernel<<<blk1(E), BT, 0, stream>>>(ew, col, deg, mm, E);
    rsqrt_kernel<<<blk1(N), BT, 0, stream>>>(deg, N);
    edge_norm_kernel<<<blk1(E), BT, 0, stream>>>(ew, row, col, deg, nrm, E);

    // ------- pre-transpose weights (contiguous B-fragment loads in WMMA GEMMs)
    transpose_kernel<<<blk1(128 * 128), BT, 0, stream>>>(Wt1, W1, HID, IN_DIM);
    transpose_kernel<<<blk1(128 * 128), BT, 0, stream>>>(Wt2, W2, HID, HID);
    transpose_kernel<<<blk1(256 * 64), BT, 0, stream>>>(Wrt, Wr, OUT_DIM, 2 * HID);

    const long nWaves = (N + 15) / 16;
    const unsigned gemmBlocks = (unsigned)((nWaves + 7) / 8);
    const unsigned scatBlocks = (unsigned)((E + 7) / 8);

    // ------- layer 1
    gemm_nt128_kernel<<<gemmBlocks, 256, 0, stream>>>(x, Wt1, hW, (int)N);
    fill_f32_kernel<<<blk1(N * 128), BT, 0, stream>>>(agg, 0.0f, N * 128);
    scatter_edges_kernel<<<scatBlocks, 256, 0, stream>>>(hW, nrm, row, col, agg, E);
    combine_relu_kernel<<<blk1(N * 32), BT, 0, stream>>>(agg, hW, deg, b1, h1, (int)N);

    // ------- layer 2 (reuse hW / agg)
    gemm_nt128_kernel<<<gemmBlocks, 256, 0, stream>>>(h1, Wt2, hW, (int)N);
    fill_f32_kernel<<<blk1(N * 128), BT, 0, stream>>>(agg, 0.0f, N * 128);
    scatter_edges_kernel<<<scatBlocks, 256, 0, stream>>>(hW, nrm, row, col, agg, E);
    combine_relu_kernel<<<blk1(N * 32), BT, 0, stream>>>(agg, hW, deg, b2, h2, (int)N);

    // ------- JK-concat readout
    gemm_readout_kernel<<<gemmBlocks, 256, 0, stream>>>(h1, h2, Wrt, br, out, (int)N);
}